// Bert_CRF_83648783057019
// MI455X (gfx1250) — compile-verified
//
#include <hip/hip_runtime.h>
#include <math.h>

typedef float v2f __attribute__((ext_vector_type(2)));
typedef float v8f __attribute__((ext_vector_type(8)));
typedef unsigned int u32x4 __attribute__((ext_vector_type(4)));
typedef int i32x8 __attribute__((ext_vector_type(8)));
typedef int i32x4 __attribute__((ext_vector_type(4)));

#define Bb 128
#define Ss 512
#define Ee 768
#define Cc 9
// TDM pads 4 dwords after every 256 dwords -> row stride 768 + 3*4 = 780 dwords.
// 780 mod 64 = 12 -> 12*r mod 64 distinct for r=0..15 -> conflict-free ds_load_b64.
#define WROW 780
#define WCHUNK 260   /* 256 data dwords + 4 pad dwords per K-chunk */

#if __has_builtin(__builtin_amdgcn_tensor_load_to_lds)
#define HAVE_TDM 1
#else
#define HAVE_TDM 0
#endif

// ---------------------------------------------------------------------------
// Kernel 1: emissions[m][c] = sum_k enc[m][k] * W[c][k] + b[c]   (m = b*S + s)
// fp32 WMMA 16x16x4, one 16x16 output tile per wave, 192 K-steps.
// W staged into LDS by the Tensor Data Mover (zero-filled rows 9..15 via OOB).
// ---------------------------------------------------------------------------
__global__ __launch_bounds__(256)
void emis_wmma(const float* __restrict__ enc, const float* __restrict__ W,
               const float* __restrict__ bias, float* __restrict__ em)
{
    __shared__ float sW[16 * WROW];
    const int tid = threadIdx.x;

#if HAVE_TDM
    if (tid < 32) {
        const unsigned long long ga = (unsigned long long)(uintptr_t)W;
        const unsigned ldsOff = (unsigned)(uintptr_t)&sW[0];
        // D# group 0: count=1, lds_addr, global_addr[56:0], type=2
        u32x4 g0;
        g0.x = 1u;
        g0.y = ldsOff;
        g0.z = (unsigned)(ga & 0xFFFFFFFFu);
        g0.w = (unsigned)((ga >> 32) & 0x01FFFFFFu) | (2u << 30);
        // D# group 1: data_size=4B(2), pad_enable=1, pad_interval=256dw(7),
        // pad_amount=4dw(3); tensor_dim0=768, tensor_dim1=9 (rows 9..15 OOB->0),
        // tile_dim0=768, tile_dim1=16, tensor_dim0_stride=768.
        i32x8 g1;
        g1[0] = (int)((2u << 16) | (1u << 20) | (7u << 22) | (3u << 25));
        g1[1] = (int)(768u << 16);   // tensor_dim0[15:0] in bits 63:48
        g1[2] = (int)(9u << 16);     // dim0 hi16=0 | tensor_dim1[15:0]=9
        g1[3] = (int)(768u << 16);   // dim1 hi16=0 | tile_dim0=768
        g1[4] = 16;                  // tile_dim1=16 | tile_dim2=0
        g1[5] = 768;                 // tensor_dim0_stride[31:0]
        g1[6] = 0;
        g1[7] = 0;
        i32x4 gz = {0, 0, 0, 0};
#if defined(__clang_major__) && __clang_major__ >= 23
        i32x8 gz8 = {0, 0, 0, 0, 0, 0, 0, 0};
        __builtin_amdgcn_tensor_load_to_lds(g0, g1, gz, gz, gz8, 0);
#else
        __builtin_amdgcn_tensor_load_to_lds(g0, g1, gz, gz, 0);
#endif
        __builtin_amdgcn_s_wait_tensorcnt(0);
    }
    __syncthreads();
#else
    // Fallback: cooperative staging into the same padded layout.
    for (int idx = tid; idx < 16 * Ee; idx += 256) {
        const int n = idx / Ee;
        const int k = idx - n * Ee;
        sW[n * WROW + k + 4 * (k >> 8)] = (n < Cc) ? W[n * Ee + k] : 0.0f;
    }
    __syncthreads();
#endif

    const int wave = tid >> 5;
    const int lane = tid & 31;
    const int half = lane >> 4;     // selects K pair {0,1} vs {2,3}
    const int r    = lane & 15;     // A: row-in-tile; B: column (class)

    const int  tile  = blockIdx.x * 8 + wave;
    const long mbase = (long)tile * 16;

    v8f acc = {0.f, 0.f, 0.f, 0.f, 0.f, 0.f, 0.f, 0.f};
    for (int kb = 0; kb < 3; ++kb) {
        const float* Ap = enc + (mbase + r) * (long)Ee + kb * 256 + 2 * half;
        const float* Bp = &sW[r * WROW + kb * WCHUNK + 2 * half];
#pragma unroll 4
        for (int k0 = 0; k0 < 256; k0 += 4) {
            v2f a  = *(const v2f*)(Ap + k0);   // K = k0+2*half, +1
            v2f bb = *(const v2f*)(Bp + k0);
            acc = __builtin_amdgcn_wmma_f32_16x16x4_f32(
                false, a, false, bb, (short)0, acc, false, false);
        }
    }

    // D layout: VGPR v -> M = v + 8*half, N = r
    if (r < Cc) {
        const float bc = bias[r];
#pragma unroll
        for (int v = 0; v < 8; ++v) {
            const long m = mbase + v + half * 8;
            em[m * Cc + r] = acc[v] + bc;
        }
    }
}

// ---------------------------------------------------------------------------
// Kernel 2: per-batch CRF. One wave / batch. Forward pass runs in the LINEAR
// domain (scaled forward algorithm): p' = (p @ 2^T2) * 2^em2, with an exact
// power-of-two renormalization (ldexp) every 8 steps, exponent tracked in off2.
// Lane c carries p[c]; broadcasts via wave32 shuffles.
// ---------------------------------------------------------------------------
__global__ __launch_bounds__(32)
void crf_fwd(const float* __restrict__ em, const float* __restrict__ start,
             const float* __restrict__ endt, const float* __restrict__ trans,
             const int* __restrict__ tags, const unsigned char* __restrict__ mask,
             float* __restrict__ partial)
{
    const int b    = blockIdx.x;
    const int lane = threadIdx.x;
    const float* emb = em   + (long)b * Ss * Cc;
    const int*   tg  = tags + (long)b * Ss;
    const unsigned char* mk = mask + (long)b * Ss;
    const float LOG2E = 1.44269504088896340736f;
    const float LN2   = 0.69314718055994530942f;

    // ---- numerator + sequence length (lane-parallel over time) ----
    float numacc = 0.f;
    float cnt    = 0.f;
    for (int t = lane; t < Ss; t += 32) {
        const bool mt = (mk[t] != 0);
        cnt += mt ? 1.f : 0.f;
        if (t >= 1 && mt)
            numacc += trans[tg[t - 1] * Cc + tg[t]] + emb[t * Cc + tg[t]];
    }
#pragma unroll
    for (int off = 16; off > 0; off >>= 1) {
        numacc += __shfl_down(numacc, off, 32);
        cnt    += __shfl_down(cnt,    off, 32);
    }

    // ---- scaled forward recurrence ----
    const int c = (lane < Cc) ? lane : (Cc - 1);  // lanes >= 9 mirror lane 8
    float Ecol[Cc];                                // 2^(T[i][c] * log2 e)
#pragma unroll
    for (int i = 0; i < Cc; ++i) Ecol[i] = exp2f(trans[i * Cc + c] * LOG2E);

    float a2 = (start[c] + emb[c]) * LOG2E;        // base-2 log alpha
    float mx0 = a2;
#pragma unroll
    for (int off = 16; off > 0; off >>= 1) mx0 = fmaxf(mx0, __shfl_xor(mx0, off, 32));
    float off2 = mx0;                              // shared base-2 exponent offset
    float p = exp2f(a2 - off2);                    // alpha = 2^off2 * p

    for (int t = 1; t < Ss; ++t) {
        const float emx = exp2f(emb[t * Cc + c] * LOG2E);  // off the alpha chain
        float pr[Cc];
#pragma unroll
        for (int i = 0; i < Cc; ++i) pr[i] = __shfl(p, i, 32) * Ecol[i];
        // balanced add tree (depth 4)
        const float s = (((pr[0] + pr[1]) + (pr[2] + pr[3])) +
                         ((pr[4] + pr[5]) + (pr[6] + pr[7]))) + pr[8];
        const float pn = s * emx;
        p = (mk[t] != 0) ? pn : p;
        if ((t & 7) == 7) {                        // exact pow2 renormalization
            float m = p;
#pragma unroll
            for (int off = 16; off > 0; off >>= 1) m = fmaxf(m, __shfl_xor(m, off, 32));
            int e;
            (void)frexpf(m, &e);                   // m = f * 2^e, f in [0.5,1)
            p = ldexpf(p, -e);                     // v_ldexp_f32: exact
            off2 += (float)e;
        }
    }

    // ---- denominator: off2 + log2(sum_c p[c] * 2^end2[c]) ----
    float se = (lane < Cc) ? p * exp2f(endt[c] * LOG2E) : 0.f;
#pragma unroll
    for (int off = 16; off > 0; off >>= 1) se += __shfl_xor(se, off, 32);
    const float den = (off2 + log2f(se)) * LN2;

    if (lane == 0) {
        const int seqlen = (int)(cnt + 0.5f);
        const float num = numacc + start[tg[0]] + emb[tg[0]] + endt[tg[seqlen - 1]];
        partial[b] = den - num;
    }
}

// ---------------------------------------------------------------------------
// Kernel 3: mean over 128 per-batch values -> scalar output
// ---------------------------------------------------------------------------
__global__ __launch_bounds__(128)
void reduce_mean(const float* __restrict__ partial, float* __restrict__ out)
{
    __shared__ float sm[128];
    const int t = threadIdx.x;
    sm[t] = partial[t];
    __syncthreads();
    for (int off = 64; off > 0; off >>= 1) {
        if (t < off) sm[t] += sm[t + off];
        __syncthreads();
    }
    if (t == 0) out[0] = sm[0] * (1.0f / Bb);
}

extern "C" void kernel_launch(void* const* d_in, const int* in_sizes, int n_in,
                              void* d_out, int out_size, void* d_ws, size_t ws_size,
                              hipStream_t stream)
{
    const float* enc  = (const float*)d_in[0];   // (B,S,E) f32
    const float* W    = (const float*)d_in[1];   // (C,E)   f32
    const float* bias = (const float*)d_in[2];   // (C,)    f32
    const float* st   = (const float*)d_in[3];   // (C,)    f32
    const float* en   = (const float*)d_in[4];   // (C,)    f32
    const float* tr   = (const float*)d_in[5];   // (C,C)   f32
    const int*   tags = (const int*)d_in[6];     // (B,S)   int
    const unsigned char* mask = (const unsigned char*)d_in[7]; // (B,S) bool

    float* em      = (float*)d_ws;                    // 128*512*9 floats
    float* partial = em + (size_t)Bb * Ss * Cc;       // 128 floats

    // 512 blocks * 8 waves * 16 rows = 65536 = B*S rows
    emis_wmma<<<512, 256, 0, stream>>>(enc, W, bias, em);
    crf_fwd<<<Bb, 32, 0, stream>>>(em, st, en, tr, tags, mask, partial);
    reduce_mean<<<1, 128, 0, stream>>>(partial, (float*)d_out);
}